// BatchAuc_49847390437818
// MI455X (gfx1250) — compile-verified
//
#include <hip/hip_runtime.h>
#include <hip/hip_bf16.h>

// ---------------------------------------------------------------------------
// Batch weighted AUC for 32 tasks x 1M examples, sort-free:
//   trap[t] = sum_{neg j} w_j * (pos weight with key > key_j  + 0.5 * tied)
// via order-preserving 18-bit binning, L2-resident histogram, WMMA-based
// suffix scan, and a gather pass. Double accumulators for the reductions.
// ---------------------------------------------------------------------------

typedef __attribute__((ext_vector_type(2))) float v2f;
typedef __attribute__((ext_vector_type(8))) float v8f;

#define BIN_BITS 18
#define NBINS (1u << BIN_BITS)

__device__ __forceinline__ unsigned mono_key(float x) {
  unsigned b = __float_as_uint(x);
  return (b & 0x80000000u) ? ~b : (b | 0x80000000u);  // monotone: larger float -> larger uint
}

__global__ __launch_bounds__(256) void zero_kernel(float4* h4, double* acc, int n4, int nd) {
  int i = blockIdx.x * blockDim.x + threadIdx.x;
  if (i < n4) h4[i] = make_float4(0.f, 0.f, 0.f, 0.f);
  if (i < nd) acc[i] = 0.0;
}

// Pass 1: per-(task,bin) positive-weight histogram + per-task pos/neg totals.
__global__ __launch_bounds__(256) void hist_kernel(
    const float* __restrict__ pred, const float* __restrict__ lab,
    const float* __restrict__ wt, float* __restrict__ hist,
    double* __restrict__ posTot, double* __restrict__ negTot, int n) {
  const int t = blockIdx.y;
  const size_t off = (size_t)t * (size_t)n;
  float* __restrict__ H = hist + (size_t)t * NBINS;
  double lp = 0.0, ln = 0.0;
  const int stride = gridDim.x * blockDim.x * 4;
  for (int i = (blockIdx.x * blockDim.x + threadIdx.x) * 4; i < n; i += stride) {
    float4 p = *(const float4*)(pred + off + i);
    float4 l = *(const float4*)(lab + off + i);
    float4 w = *(const float4*)(wt + off + i);
    float pv[4] = {p.x, p.y, p.z, p.w};
    float lv[4] = {l.x, l.y, l.z, l.w};
    float wv[4] = {w.x, w.y, w.z, w.w};
#pragma unroll
    for (int k = 0; k < 4; ++k) {
      unsigned bin = mono_key(pv[k]) >> (32 - BIN_BITS);
      float pw = wv[k] * lv[k];                 // labels are exactly 0.0 / 1.0
      lp += (double)pw;
      ln += (double)(wv[k] - pw);
      if (pw != 0.f) atomicAdd(&H[bin], pw);    // L2-resident table (32MB << 192MB)
    }
  }
  __shared__ double sp[256], sn[256];
  int tid = threadIdx.x;
  sp[tid] = lp; sn[tid] = ln;
  __syncthreads();
  for (int s = 128; s > 0; s >>= 1) {
    if (tid < s) { sp[tid] += sp[tid + s]; sn[tid] += sn[tid + s]; }
    __syncthreads();
  }
  if (tid == 0) { atomicAdd(&posTot[t], sp[0]); atomicAdd(&negTot[t], sn[0]); }
}

// Suffix scan over NBINS per task; overwrite hist[b] with
//   G[b] = (inclusive suffix sum at b) - 0.5*hist[b]  = posAbove(b) + 0.5*posSame(b)
// One workgroup (8 waves) per task. Each wave scans 256 bins (reversed order)
// using V_WMMA_F32_16X16X4_F32: Y = L16 x X (K=16 via 4 chained K=4 WMMAs)
// gives within-column inclusive prefixes; column/ wave/ chunk offsets via
// lane shuffles and LDS carry.
__global__ __launch_bounds__(256) void scan_kernel(float* __restrict__ hist) {
  const int t = blockIdx.x;
  float* __restrict__ H = hist + (size_t)t * NBINS;
  __shared__ float stage[2048];
  __shared__ float wsum[8];
  __shared__ float carryS;
  const int tid = threadIdx.x;
  const int wid = tid >> 5;
  const int lane = tid & 31;
  const int n = lane & 15;     // column index (N) for B/C/D operands
  const int half = lane >> 4;  // which half-wave
  if (tid == 0) carryS = 0.f;
  __syncthreads();

  for (unsigned base = 0; base < NBINS; base += 2048) {
    // Load 2048 bins in reversed order: stage[p] = H[NBINS-1 - (base+p)]
#pragma unroll
    for (int q = 0; q < 8; ++q) {
      int p = (wid << 8) + (q << 5) + lane;
      stage[p] = H[(NBINS - 1u) - (base + (unsigned)p)];
    }
    __syncthreads();

    // --- per-wave 256-element inclusive scan via WMMA f32 16x16x4 ---
    // Element p (0..255) sits at X[m][n0] with m=p%16, n0=p/16.
    // Y = Lfull * X : Y[m][n0] = sum_{m'<=m} X[m'][n0]  (within-column prefix)
    v8f acc = {0.f, 0.f, 0.f, 0.f, 0.f, 0.f, 0.f, 0.f};
    const float* xs = &stage[wid << 8];
#pragma unroll
    for (int c = 0; c < 4; ++c) {
      v2f a, b;
      // A (16x4 chunk of lower-triangular ones): lane M = lane%16,
      // VGPR j holds K = 2*(lane/16)+j  (ISA 7.12.2, 32-bit A 16x4 layout)
      int ka = 4 * c + 2 * half;
      a.x = (ka <= n) ? 1.f : 0.f;
      a.y = (ka + 1 <= n) ? 1.f : 0.f;
      // B (4x16 chunk of X): row striped across lanes; lane N = lane%16,
      // VGPR j holds K = (lane/16) + 2j
      int kb = 4 * c + half;
      b.x = xs[(n << 4) + kb];
      b.y = xs[(n << 4) + kb + 2];
      acc = __builtin_amdgcn_wmma_f32_16x16x4_f32(
          false, a, false, b, (short)0, acc, false, false);
    }
    // Column totals: Y[15][n] lives in VGPR7 of lane 16+n.
    float cs = __shfl(acc[7], 16 + n, 32);
    // Exclusive scan of column totals across the 16 columns (both halves dup).
    float incl = cs;
#pragma unroll
    for (int d = 1; d < 16; d <<= 1) {
      float u = __shfl_up(incl, (unsigned)d, 16);
      if (n >= d) incl += u;
    }
    float excl = incl - cs;
    float wtot = __shfl(incl, 15, 16);
    if (lane == 0) wsum[wid] = wtot;
    __syncthreads();

    float woff = carryS;
    for (int i2 = 0; i2 < wid; ++i2) woff += wsum[i2];

    // D layout: VGPR r holds (M = r + 8*half, N = n). p = n*16 + M.
#pragma unroll
    for (int r = 0; r < 8; ++r) {
      int M = r + (half << 3);
      int p = (n << 4) + M;
      float sinc = acc[r] + excl + woff;  // inclusive suffix sum at this bin
      unsigned binIdx = (NBINS - 1u) - (base + (unsigned)((wid << 8) + p));
      H[binIdx] = sinc - 0.5f * xs[p];    // posAbove + 0.5*posSame
    }
    __syncthreads();
    if (tid == 0) {
      float s = 0.f;
      for (int i2 = 0; i2 < 8; ++i2) s += wsum[i2];
      carryS += s;
    }
    __syncthreads();
  }
}

// Pass 2: trap[t] += sum over negatives of w * G[bin]  (gathers hit L2).
__global__ __launch_bounds__(256) void pass2_kernel(
    const float* __restrict__ pred, const float* __restrict__ lab,
    const float* __restrict__ wt, const float* __restrict__ hist,
    double* __restrict__ trap, int n) {
  const int t = blockIdx.y;
  const size_t off = (size_t)t * (size_t)n;
  const float* __restrict__ G = hist + (size_t)t * NBINS;
  double s = 0.0;
  const int stride = gridDim.x * blockDim.x * 4;
  for (int i = (blockIdx.x * blockDim.x + threadIdx.x) * 4; i < n; i += stride) {
    float4 p = *(const float4*)(pred + off + i);
    float4 l = *(const float4*)(lab + off + i);
    float4 w = *(const float4*)(wt + off + i);
    float pv[4] = {p.x, p.y, p.z, p.w};
    float lv[4] = {l.x, l.y, l.z, l.w};
    float wv[4] = {w.x, w.y, w.z, w.w};
#pragma unroll
    for (int k = 0; k < 4; ++k) {
      float wn = wv[k] * (1.f - lv[k]);
      if (wn != 0.f) {
        unsigned bin = mono_key(pv[k]) >> (32 - BIN_BITS);
        s += (double)wn * (double)G[bin];
      }
    }
  }
  __shared__ double sd[256];
  int tid = threadIdx.x;
  sd[tid] = s;
  __syncthreads();
  for (int st = 128; st > 0; st >>= 1) {
    if (tid < st) sd[tid] += sd[tid + st];
    __syncthreads();
  }
  if (tid == 0) atomicAdd(&trap[t], sd[0]);
}

__global__ void final_kernel(const double* __restrict__ trap,
                             const double* __restrict__ posTot,
                             const double* __restrict__ negTot,
                             float* __restrict__ out, int T) {
  int t = blockIdx.x * blockDim.x + threadIdx.x;
  if (t < T) {
    double fac = posTot[t] * negTot[t];
    out[t] = (fac == 0.0) ? 0.5f : (float)(trap[t] / fac);
  }
}

extern "C" void kernel_launch(void* const* d_in, const int* in_sizes, int n_in,
                              void* d_out, int out_size, void* d_ws, size_t ws_size,
                              hipStream_t stream) {
  // inputs: [0]=n_tasks scalar, [1]=predictions, [2]=labels, [3]=weights
  const float* pred = (const float*)d_in[1];
  const float* lab = (const float*)d_in[2];
  const float* wt = (const float*)d_in[3];
  const int T = out_size;              // 32 tasks
  const int n = in_sizes[1] / T;       // 1,000,000 examples per task

  // workspace: [ T*NBINS float hist | T doubles trap | T posTot | T negTot ]
  float* hist = (float*)d_ws;
  double* acc = (double*)((char*)d_ws + (size_t)T * NBINS * sizeof(float));
  double* trap = acc;
  double* posTot = acc + T;
  double* negTot = acc + 2 * T;

  int n4 = (int)(((size_t)T * NBINS) / 4);
  int nd = 3 * T;
  int zmax = (n4 > nd) ? n4 : nd;
  zero_kernel<<<(zmax + 255) / 256, 256, 0, stream>>>((float4*)hist, acc, n4, nd);

  dim3 hgrid((unsigned)((n + 256 * 4 - 1) / (256 * 4)), (unsigned)T);
  hist_kernel<<<hgrid, 256, 0, stream>>>(pred, lab, wt, hist, posTot, negTot, n);
  scan_kernel<<<T, 256, 0, stream>>>(hist);
  pass2_kernel<<<hgrid, 256, 0, stream>>>(pred, lab, wt, hist, trap, n);
  final_kernel<<<(T + 63) / 64, 64, 0, stream>>>(trap, posTot, negTot, (float*)d_out, T);
}